// ConvCNP_30657476558904
// MI455X (gfx1250) — compile-verified
//
#include <hip/hip_runtime.h>

typedef __attribute__((ext_vector_type(2))) float v2f;
typedef __attribute__((ext_vector_type(8))) float v8f;

#define N_CTX   2048
#define M_TGT   8192
#define N_BATCH 4
#define C_OUT   64
#define ROWS    256           // rows (m values) per workgroup
#define EPS_F   1e-8f

__global__ __launch_bounds__(ROWS)
void convcnp_fused_kernel(const float* __restrict__ context_in,   // (B,N,1)
                          const float* __restrict__ context_out,  // (B,N,1)
                          const float* __restrict__ target_in,    // (B,M,1)
                          const float* __restrict__ sigma,        // (2,)
                          const float* __restrict__ W,            // (64,2) row-major
                          const float* __restrict__ bias,         // (64,)
                          float* __restrict__ out)                // (B,M,64)
{
    __shared__ float2 ctx[N_CTX];   // (context_in, context_out) interleaved
    __shared__ float2 sd[ROWS];     // (density, conv) staging for WMMA

    const int tid = threadIdx.x;
    const int blk = blockIdx.x;
    const int b   = blk >> 5;            // 32 blocks per batch (8192/256)
    const int m0  = (blk & 31) << 8;     // first m of this block

    // ---- stage context for this batch into LDS (reused 256x by this block) ----
    const float* ci = context_in  + b * N_CTX;
    const float* co = context_out + b * N_CTX;
    #pragma unroll
    for (int i = 0; i < N_CTX / ROWS; ++i) {
        int n = tid + i * ROWS;
        ctx[n] = make_float2(ci[n], co[n]);
    }

    // fold -0.5/scale^2 into a log2(e)-scaled constant: exp(x*k') = exp2(x*k)
    const float log2e = 1.44269504088896340736f;
    float sc0 = __expf(sigma[0]);
    float sc1 = __expf(sigma[1]);
    float k0 = -0.5f * log2e / (sc0 * sc0);
    float k1 = -0.5f * log2e / (sc1 * sc1);

    __syncthreads();

    // ---- stage 1: per-thread Gaussian reduction over all context points ----
    const int   m = m0 + tid;
    const float t = target_in[(size_t)b * M_TGT + m];

    float s0 = 0.f, s1 = 0.f;
    #pragma unroll 8
    for (int n = 0; n < N_CTX; ++n) {
        float2 cc = ctx[n];                 // ds_load_b64, same addr in wave -> broadcast
        float d  = cc.x - t;
        float d2 = d * d;
        s0 += __builtin_amdgcn_exp2f(d2 * k0);           // v_exp_f32
        s1 += cc.y * __builtin_amdgcn_exp2f(d2 * k1);    // v_exp_f32 + fma
    }
    float density = s0;
    float conv    = s1 / (s0 + EPS_F);

    sd[tid] = make_float2(density, conv);
    __syncthreads();

    // ---- stage 2: (256x2) @ (2x64) + bias via V_WMMA_F32_16X16X4_F32 ----
    // A (16x4 f32): lanes 0-15 hold M=lane, VGPR0=K0(density), VGPR1=K1(conv);
    //               lanes 16-31 hold K=2,3 -> zero padding.
    // B (4x16 f32): lanes 0-15 hold N=lane, VGPR0=W[col][0], VGPR1=W[col][1];
    //               lanes 16-31 (K=2,3) zero.
    // C/D (16x16 f32, 8 VGPRs): VGPR r = row r (lanes 0-15) / row r+8 (lanes 16-31).
    const int  wave    = tid >> 5;
    const int  lane    = tid & 31;
    const int  lo      = lane & 15;
    const bool lowHalf = (lane < 16);

    const size_t rowBaseG = (size_t)b * M_TGT + m0;   // global bm row of this block

    #pragma unroll
    for (int rt = 0; rt < 2; ++rt) {
        const int rowBase = wave * 32 + rt * 16;
        v2f a;
        if (lowHalf) {
            float2 v = sd[rowBase + lo];
            a.x = v.x; a.y = v.y;
        } else {
            a.x = 0.f; a.y = 0.f;
        }
        #pragma unroll
        for (int jt = 0; jt < 4; ++jt) {
            const int col = jt * 16 + lo;
            v2f bb;
            if (lowHalf) { bb.x = W[2 * col]; bb.y = W[2 * col + 1]; }
            else         { bb.x = 0.f;        bb.y = 0.f; }

            const float cb = bias[col];
            v8f c = { cb, cb, cb, cb, cb, cb, cb, cb };

            v8f d = __builtin_amdgcn_wmma_f32_16x16x4_f32(
                        /*neg_a=*/false, a, /*neg_b=*/false, bb,
                        /*c_mod=*/(short)0, c, /*reuse_a=*/false, /*reuse_b=*/false);

            const int rOff = lowHalf ? 0 : 8;
            #pragma unroll
            for (int r = 0; r < 8; ++r) {
                size_t row = rowBaseG + (size_t)(rowBase + r + rOff);
                out[row * C_OUT + col] = d[r];
            }
        }
    }
}

extern "C" void kernel_launch(void* const* d_in, const int* in_sizes, int n_in,
                              void* d_out, int out_size, void* d_ws, size_t ws_size,
                              hipStream_t stream) {
    (void)in_sizes; (void)n_in; (void)out_size; (void)d_ws; (void)ws_size;
    const float* context_in  = (const float*)d_in[0];
    const float* context_out = (const float*)d_in[1];
    const float* target_in   = (const float*)d_in[2];
    const float* sigma       = (const float*)d_in[3];
    const float* W           = (const float*)d_in[4];
    const float* bias        = (const float*)d_in[5];
    float*       out         = (float*)d_out;

    const int grid = (N_BATCH * M_TGT) / ROWS;   // 128 workgroups
    convcnp_fused_kernel<<<grid, ROWS, 0, stream>>>(
        context_in, context_out, target_in, sigma, W, bias, out);
}